// scintegral_model_12549894439584
// MI455X (gfx1250) — compile-verified
//
#include <hip/hip_runtime.h>
#include <math.h>

#define N_SAMPLE 50000
#define N_GENE   128
#define N_TYPE   16
#define N_BATCH  4

typedef float v2f __attribute__((ext_vector_type(2)));
typedef float v8f __attribute__((ext_vector_type(8)));

// ---- workspace layout (float offsets) ----
#define OFF_D    0                    // 128*16  : D[g][t] = marker*delta
#define OFF_E    2048                 // 128*16  : E[g][t] = exp(D)
#define OFF_PHI  4096                 // 128     : clamped phi
#define OFF_T    4224                 // 128*20  : T[g][k] = lg(phi+k)-lg(phi)-lg(1+k)
#define OFF_C0   6784                 // 1       : sum_g phi*log(phi)
#define OFF_LSE  6912                 // N_SAMPLE: per-sample logsumexp_t(B)
#define OFF_BLK  (OFF_LSE + N_SAMPLE) // per-block partial sums

#define LN2F 0.69314718055994531f

// ------------------------------------------------------------------
// Kernel 1: per-gene tables (1 block, 128 threads, one gene each)
// ------------------------------------------------------------------
__global__ __launch_bounds__(128) void setup_tables(
    const float* __restrict__ delta, const float* __restrict__ phi,
    const float* __restrict__ marker, float* __restrict__ ws) {
  const int g = threadIdx.x;           // 0..127
  float p = fminf(fmaxf(phi[g], 1.0f), 100.0f);
  ws[OFF_PHI + g] = p;
  for (int t = 0; t < N_TYPE; ++t) {
    float d = marker[g * N_TYPE + t] * delta[g * N_TYPE + t];
    ws[OFF_D + g * N_TYPE + t] = d;
    ws[OFF_E + g * N_TYPE + t] = __expf(d);
  }
  // T[g][k] = sum_{j<k} log(p+j) - log(k!)
  float run = 0.0f, lfac = 0.0f;
  for (int k = 0; k < 20; ++k) {
    ws[OFF_T + g * 20 + k] = run - lfac;
    run  += __logf(p + (float)k);
    lfac += __logf((float)(k + 1));
  }
  __shared__ float red[128];
  red[g] = p * __logf(p);
  __syncthreads();
  for (int s = 64; s > 0; s >>= 1) {
    if (g < s) red[g] += red[g + s];
    __syncthreads();
  }
  if (g == 0) ws[OFF_C0] = red[0];
}

// ------------------------------------------------------------------
// Kernel 2: fused WMMA (eta GEMM K=4, kD GEMM K=128) + NB log loop
//           + per-sample logsumexp over 16 types.
// Block = 128 threads = 4 waves; each wave owns a 16-sample tile.
// ------------------------------------------------------------------
__global__ __launch_bounds__(128) void nb_lse_kernel(
    const float* __restrict__ beta,  // [4][128]
    const float* __restrict__ expr,  // [S][128]
    const float* __restrict__ cov,   // [S][4]
    const float* __restrict__ sf,    // [S]
    const float* __restrict__ ws,
    float* __restrict__ lse_out) {
  __shared__ __align__(16) float         m_lds[4][16][132]; // 33792 B (132-pad: b128 rows tile 64 banks)
  __shared__ __align__(16) unsigned char k_lds[4][16][132]; //  8448 B (k as u8, exact for 0..19)
  __shared__ __align__(16) float         EP[128][16];       //  8192 B: E[g][t], 64B rows
  __shared__ __align__(16) float         Dl[128][16];       //  8192 B: D[g][t]
  __shared__ __align__(16) float         phil[128];         //   512 B: clamped phi

  const int tid  = threadIdx.x;
  const int wave = tid >> 5;
  const int lane = tid & 31;
  const int h    = lane >> 4;   // lane half
  const int sl   = lane & 15;

  // cooperative table loads
  for (int i = tid; i < 2048; i += 128) {
    EP[i >> 4][i & 15] = ws[OFF_E + i];
    Dl[i >> 4][i & 15] = ws[OFF_D + i];
  }
  phil[tid] = ws[OFF_PHI + tid];

  const int s0 = blockIdx.x * 64 + wave * 16;

  // k tile (coalesced: lanes sweep gene dim)
  for (int i = lane; i < 2048; i += 32) {
    const int g  = i & 127;
    const int sr = i >> 7;
    const int s  = min(s0 + sr, N_SAMPLE - 1);
    k_lds[wave][sr][g] = (unsigned char)expr[s * N_GENE + g];
  }
  __syncthreads();

  // ---- eta GEMM: C[M=sample][N=gene] = cov(16x4) x beta(4x16) per tile ----
  // A 16x4 f32 layout: lanes 0-15 K=0,1 ; lanes 16-31 K=2,3 ; M = lane&15
  const int sA = min(s0 + sl, N_SAMPLE - 1);
  v2f a_cov;
  a_cov.x = cov[sA * N_BATCH + 2 * h];
  a_cov.y = cov[sA * N_BATCH + 2 * h + 1];
  float sfv[8];
#pragma unroll
  for (int v = 0; v < 8; ++v)
    sfv[v] = sf[min(s0 + v + 8 * h, N_SAMPLE - 1)];

  for (int gt = 0; gt < 8; ++gt) {
    const int g = gt * 16 + sl;                    // C col N = lane&15
    v2f b;
    b.x = beta[(2 * h) * N_GENE + g];              // B 4x16: K split across halves
    b.y = beta[(2 * h + 1) * N_GENE + g];
    v8f c = {};
    c = __builtin_amdgcn_wmma_f32_16x16x4_f32(false, a_cov, false, b,
                                              (short)0, c, false, false);
#pragma unroll
    for (int v = 0; v < 8; ++v)                    // C row M = v + 8*h
      m_lds[wave][v + 8 * h][g] = sfv[v] * __expf(c[v]);
  }

  // ---- kD GEMM: C[M=type][N=sample] = D^T(16xK) x k(Kx16), K=128 as 32x K=4 ----
  v8f acc = {};
  for (int c4 = 0; c4 < 32; ++c4) {
    const int g0 = c4 * 4;
    v2f a, b;
    a.x = Dl[g0 + 2 * h][sl];                      // A: M=t=lane&15, K per half
    a.y = Dl[g0 + 2 * h + 1][sl];
    b.x = (float)k_lds[wave][sl][g0 + 2 * h];      // B: N=s=lane&15, K per half
    b.y = (float)k_lds[wave][sl][g0 + 2 * h + 1];
    acc = __builtin_amdgcn_wmma_f32_16x16x4_f32(false, a, false, b,
                                                (short)0, acc, false, false);
  }

  // ---- heavy loop: acc[v] -= (k+phi) * ln(m*E[g][t] + phi) ----
  // lane owns sample sl, types t = 8*h + v (matches WMMA C layout).
  // Vectorized LDS: b128 for m/phi/E rows, b32 for 4 packed k bytes.
  const int tb = 8 * h;
  const float*         mrow = &m_lds[wave][sl][0];
  const unsigned char* krow = &k_lds[wave][sl][0];
  for (int g = 0; g < N_GENE; g += 4) {
    const float4 mv4 = *(const float4*)(mrow + g);
    const unsigned int kb = *(const unsigned int*)(krow + g);
    const float4 ph4 = *(const float4*)(&phil[g]);
    const float mv[4] = {mv4.x, mv4.y, mv4.z, mv4.w};
    const float ph[4] = {ph4.x, ph4.y, ph4.z, ph4.w};
#pragma unroll
    for (int j = 0; j < 4; ++j) {
      const float kv = (float)((kb >> (8 * j)) & 0xffu); // v_cvt_f32_ubyteN
      const float w  = (kv + ph[j]) * LN2F;              // fold ln2 into coeff
      const float4 e0 = *(const float4*)(&EP[g + j][tb]);
      const float4 e1 = *(const float4*)(&EP[g + j][tb + 4]);
      const float e[8] = {e0.x, e0.y, e0.z, e0.w, e1.x, e1.y, e1.z, e1.w};
#pragma unroll
      for (int v = 0; v < 8; ++v) {
        const float x = fmaf(mv[j], e[v], ph[j]);
        acc[v] = fmaf(-w, __log2f(x), acc[v]);           // v_log_f32 (TRANS)
      }
    }
  }

  // ---- logsumexp over 16 types (8 local + partner lane via shfl_xor 16) ----
  float mx = acc[0];
#pragma unroll
  for (int v = 1; v < 8; ++v) mx = fmaxf(mx, acc[v]);
  mx = fmaxf(mx, __shfl_xor(mx, 16, 32));
  float se = 0.0f;
#pragma unroll
  for (int v = 0; v < 8; ++v) se += __expf(acc[v] - mx);
  se += __shfl_xor(se, 16, 32);
  if (h == 0 && (s0 + sl) < N_SAMPLE)
    lse_out[s0 + sl] = mx + LN2F * __log2f(se);
}

// ------------------------------------------------------------------
// Kernel 3: t-independent per-sample terms A_s + block partial sums
// ------------------------------------------------------------------
__global__ __launch_bounds__(256) void persample_kernel(
    const float* __restrict__ beta, const float* __restrict__ expr,
    const float* __restrict__ cov, const float* __restrict__ sf,
    const float* __restrict__ ws, float* __restrict__ blk) {
  __shared__ float Tl[2560];
  __shared__ float wred[8];
  const int tid = threadIdx.x;
  for (int i = tid; i < 2560; i += 256) Tl[i] = ws[OFF_T + i];
  __syncthreads();

  const int s  = blockIdx.x * 256 + tid;
  const int sc = min(s, N_SAMPLE - 1);
  const float c0 = cov[sc * 4 + 0], c1 = cov[sc * 4 + 1];
  const float c2 = cov[sc * 4 + 2], c3 = cov[sc * 4 + 3];
  float accT = 0.0f, keta = 0.0f, Ks = 0.0f;
  for (int g = 0; g < N_GENE; ++g) {
    const float kf  = expr[sc * N_GENE + g];
    const float eta = c0 * beta[g] + c1 * beta[128 + g] +
                      c2 * beta[256 + g] + c3 * beta[384 + g];
    accT += Tl[g * 20 + (int)kf];
    keta  = fmaf(kf, eta, keta);
    Ks   += kf;
  }
  const float A   = accT + keta + __logf(sf[sc]) * Ks + ws[OFF_C0];
  float tot = (s < N_SAMPLE) ? (A + ws[OFF_LSE + s]) : 0.0f;

  for (int o = 16; o > 0; o >>= 1) tot += __shfl_down(tot, o, 32);
  if ((tid & 31) == 0) wred[tid >> 5] = tot;
  __syncthreads();
  if (tid == 0) {
    float bsum = 0.0f;
    for (int w = 0; w < 8; ++w) bsum += wred[w];
    blk[blockIdx.x] = bsum;
  }
}

// Deterministic final reduction (fixed order -> replay-stable)
__global__ void final_reduce(const float* __restrict__ blk,
                             float* __restrict__ out, int n) {
  if (threadIdx.x == 0) {
    float s = 0.0f;
    for (int i = 0; i < n; ++i) s += blk[i];
    out[0] = s;
  }
}

// ------------------------------------------------------------------
extern "C" void kernel_launch(void* const* d_in, const int* in_sizes, int n_in,
                              void* d_out, int out_size, void* d_ws, size_t ws_size,
                              hipStream_t stream) {
  (void)in_sizes; (void)n_in; (void)out_size; (void)ws_size;
  const float* delta  = (const float*)d_in[0];
  const float* beta   = (const float*)d_in[1];
  const float* phi    = (const float*)d_in[2];
  const float* expr   = (const float*)d_in[3];
  const float* cov    = (const float*)d_in[4];
  const float* sf     = (const float*)d_in[5];
  const float* marker = (const float*)d_in[6];
  float* ws  = (float*)d_ws;
  float* out = (float*)d_out;

  setup_tables<<<1, 128, 0, stream>>>(delta, phi, marker, ws);

  const int b2 = (N_SAMPLE + 63) / 64;   // 4 waves x 16 samples per block
  nb_lse_kernel<<<b2, 128, 0, stream>>>(beta, expr, cov, sf, ws, ws + OFF_LSE);

  const int b3 = (N_SAMPLE + 255) / 256;
  persample_kernel<<<b3, 256, 0, stream>>>(beta, expr, cov, sf, ws, ws + OFF_BLK);

  final_reduce<<<1, 32, 0, stream>>>(ws + OFF_BLK, out, b3);
}